// upc_sdg_83992380440899
// MI455X (gfx1250) — compile-verified
//
#include <hip/hip_runtime.h>

typedef float v2f __attribute__((ext_vector_type(2)));
typedef float v8f __attribute__((ext_vector_type(8)));

#define B_    32
#define L_    50
#define D_    64
#define NI_   50000
#define ROWS  1600          // B*L
#define NCHUNK 125
#define TILES_PER_CHUNK 25  // 125*25 = 3125 = NI/16 column tiles
#define BITS_W 1568         // ceil(50000/32)=1563, padded to multiple of 32

// ---------------- blocked-item bitmask ----------------
__global__ void bits_zero_kernel(unsigned* __restrict__ bits) {
  int t = blockIdx.x * 256 + threadIdx.x;
  if (t < B_ * BITS_W) bits[t] = 0u;
}

__global__ void bits_scatter_kernel(const int* __restrict__ inter_idx,
                                    const unsigned char* __restrict__ mask,
                                    unsigned* __restrict__ bits) {
  int t = blockIdx.x * blockDim.x + threadIdx.x;
  if (t < ROWS && mask[t]) {
    int b = t / L_;
    int idx = inter_idx[t];
    atomicOr(&bits[b * BITS_W + (idx >> 5)], 1u << (idx & 31));
  }
}

// ---------------- gathers + attention MLP + gate MLP ----------------
__global__ __launch_bounds__(64) void mlp_kernel(
    const int* __restrict__ user_idx, const int* __restrict__ inter_idx,
    const float* __restrict__ priv, const unsigned char* __restrict__ mask,
    const float* __restrict__ user_table, const float* __restrict__ item_table,
    const float* __restrict__ w1, const float* __restrict__ b1,
    const float* __restrict__ w2, const float* __restrict__ b2,
    const float* __restrict__ wg, const float* __restrict__ bg,
    float* __restrict__ out_user, float* __restrict__ out_item,
    float* __restrict__ ws_latent, float* __restrict__ ws_att)
{
  __shared__ float su[D_], si[D_], red[D_];
  const int row = blockIdx.x;
  const int b = row / L_, l = row % L_;
  const int d = threadIdx.x;

  const int ui = user_idx[b];
  const int ii = inter_idx[row];
  su[d] = user_table[(size_t)ui * D_ + d];
  si[d] = item_table[(size_t)ii * D_ + d];
  __syncthreads();

  float hacc = b1[d];
  float lacc = bg[d];
  for (int k = 0; k < D_; ++k) {
    float u = su[k];
    hacc = fmaf(u, w1[k * D_ + d], hacc);
    lacc = fmaf(u, wg[k * D_ + d], lacc);
  }
  for (int k = 0; k < D_; ++k) {
    float x = si[k];
    hacc = fmaf(x, w1[(D_ + k) * D_ + d], hacc);
    lacc = fmaf(x, wg[(D_ + k) * D_ + d], lacc);
  }
  lacc = fmaf(priv[b], wg[(2 * D_) * D_ + d], lacc);

  ws_latent[(size_t)row * D_ + d] = lacc;
  out_item[(size_t)row * D_ + d] = si[d];
  if (l == 0) out_user[b * D_ + d] = su[d];

  red[d] = hacc * w2[d];
  __syncthreads();
  for (int s = 32; s > 0; s >>= 1) {
    if (d < s) red[d] += red[d + s];
    __syncthreads();
  }
  if (d == 0) {
    float a = red[0] + b2[0];
    a = (a >= 0.f) ? a : 0.01f * a;               // LeakyReLU
    ws_att[row] = mask[row] ? a : -__builtin_inff();
  }
}

// ---------------- softmax over L per batch ----------------
__global__ __launch_bounds__(64) void softmax_kernel(const float* __restrict__ att,
                                                     float* __restrict__ out) {
  __shared__ float red[64];
  __shared__ float smax, ssum;
  const int b = blockIdx.x, t = threadIdx.x;
  float a = (t < L_) ? att[b * L_ + t] : -__builtin_inff();
  red[t] = a; __syncthreads();
  for (int s = 32; s > 0; s >>= 1) {
    if (t < s) red[t] = fmaxf(red[t], red[t + s]);
    __syncthreads();
  }
  if (t == 0) smax = red[0];
  __syncthreads();
  float e = (t < L_) ? expf(a - smax) : 0.f;
  red[t] = e; __syncthreads();
  for (int s = 32; s > 0; s >>= 1) {
    if (t < s) red[t] += red[t + s];
    __syncthreads();
  }
  if (t == 0) ssum = red[0];
  __syncthreads();
  if (t < L_) out[b * L_ + t] = e / ssum;
}

// ---------------- fused dual WMMA GEMM + gumbel argmax ----------------
// Workgroup = 4 waves; each wave owns one 16-row tile (dist + sim chains).
// The 4KB B tile (16 items x 64 K, f32) is loaded ONCE per workgroup into
// LDS (double-buffered, fragment-major layout) and shared by all 4 waves:
//   frag(kk,h,n) = kk*32 + h*16 + n  (v2f)   -> conflict-free ds_load_b64
// A layout (V_WMMA_F32_16X16X4_F32): lane (m=lane&15, h=lane>>4) holds
// A[m][4kk+2h], A[m][4kk+2h+1] per k-step kk. C/D: VGPR j = row j+8h, col lane&15.
__global__ __launch_bounds__(128) void gemm_kernel(
    const float* __restrict__ item_table,
    const int* __restrict__ inter_idx,
    const float* __restrict__ latent,
    const float* __restrict__ gumbel,
    const unsigned* __restrict__ bits,
    float* __restrict__ out_dist,
    float* __restrict__ pval,
    int* __restrict__ pidx)
{
  __shared__ v2f smem[2][512];   // two 4 KB B-tile buffers

  const int tid = threadIdx.x;
  const int wv = tid >> 5;       // wave 0..3
  const int lane = tid & 31;
  const int n = lane & 15;
  const int h = lane >> 4;

  const int rowTile = blockIdx.x * 4 + wv;             // 0..99
  const int chunk = blockIdx.y;                        // 0..124
  const int colBase = chunk * (TILES_PER_CHUNK * 16);

  // ---- A operands for this wave's row-tile (K=64 -> 16 k-steps) ----
  const int arow = rowTile * 16 + n;
  const int aitem = inter_idx[arow];
  const float* ad = item_table + (size_t)aitem * D_ + 2 * h;
  const float* al = latent + (size_t)arow * D_ + 2 * h;
  v2f aD[16], aL[16];
#pragma unroll
  for (int kk = 0; kk < 16; ++kk) {
    aD[kk] = *(const v2f*)(ad + 4 * kk);
    aL[kk] = *(const v2f*)(al + 4 * kk);
  }

  // per-lane output rows: r = rl + j, j = 0..7
  const int rl = rowTile * 16 + 8 * h;
  const size_t obase = (size_t)rl * NI_;
  // a 16-row tile spans at most 2 batches
  const int bA = (rowTile * 16) / L_;
  const int bB = (rowTile * 16 + 15) / L_;
  const int bndry = (bA + 1) * L_;

  float best[8]; int bidx[8];
#pragma unroll
  for (int j = 0; j < 8; ++j) { best[j] = -__builtin_inff(); bidx[j] = 0x3fffffff; }

  // cooperative B-tile fill: 128 threads, thread = (fn row, fq kk-pair)
  const int fn = tid & 15;
  const int fq = tid >> 4;       // 0..7 -> kk = 2fq, 2fq+1
  auto fill = [&](int buf, int col0) {
    const float* rp = item_table + (size_t)(col0 + fn) * D_ + fq * 8;
    float4 fa = *(const float4*)(rp);
    float4 fb = *(const float4*)(rp + 4);
    v2f* s = smem[buf];
    v2f v0; v0.x = fa.x; v0.y = fa.y;
    v2f v1; v1.x = fa.z; v1.y = fa.w;
    v2f v2; v2.x = fb.x; v2.y = fb.y;
    v2f v3; v3.x = fb.z; v3.y = fb.w;
    s[(2 * fq) * 32 + fn]          = v0;   // frag(2fq,   h=0, fn)
    s[(2 * fq) * 32 + 16 + fn]     = v1;   // frag(2fq,   h=1, fn)
    s[(2 * fq + 1) * 32 + fn]      = v2;   // frag(2fq+1, h=0, fn)
    s[(2 * fq + 1) * 32 + 16 + fn] = v3;   // frag(2fq+1, h=1, fn)
  };

  fill(0, colBase);
  __syncthreads();

  for (int t = 0; t < TILES_PER_CHUNK; ++t) {
    const int cur = t & 1;
    if (t + 1 < TILES_PER_CHUNK) fill(cur ^ 1, colBase + (t + 1) * 16);

    const v2f* bs = &smem[cur][h * 16 + n];
    v8f cD = {0.f, 0.f, 0.f, 0.f, 0.f, 0.f, 0.f, 0.f};
    v8f cS = {0.f, 0.f, 0.f, 0.f, 0.f, 0.f, 0.f, 0.f};
#pragma unroll
    for (int kk = 0; kk < 16; ++kk) {
      v2f bv = bs[kk * 32];                // conflict-free ds_load_b64
      cD = __builtin_amdgcn_wmma_f32_16x16x4_f32(false, aD[kk], false, bv,
                                                 (short)0, cD, false, false);
      cS = __builtin_amdgcn_wmma_f32_16x16x4_f32(false, aL[kk], false, bv,
                                                 (short)0, cS, false, false);
    }

    const int col = colBase + t * 16 + n;
    const unsigned word_off = (unsigned)col >> 5;
    const unsigned sh = (unsigned)col & 31u;
    const unsigned wAv = bits[bA * BITS_W + word_off];
    const unsigned wBv = bits[bB * BITS_W + word_off];
    float* dp = out_dist + obase + (size_t)col;
    const float* gp = gumbel + obase + (size_t)col;
#pragma unroll
    for (int j = 0; j < 8; ++j) {
      __builtin_nontemporal_store(cD[j], dp + (size_t)j * NI_);   // imm offset
      float g = __builtin_nontemporal_load(gp + (size_t)j * NI_);
      unsigned w = ((rl + j) < bndry) ? wAv : wBv;
      float logit = ((w >> sh) & 1u) ? -__builtin_inff() : (cS[j] + g);
      if (logit > best[j]) { best[j] = logit; bidx[j] = col; }
    }
    __syncthreads();   // all reads of smem[cur] done; fill of cur^1 visible next iter
  }

  // argmax reduce across the 16 lanes of each half (xor masks stay in-half).
#pragma unroll
  for (int j = 0; j < 8; ++j) {
    float v = best[j]; int i = bidx[j];
    for (int off = 8; off >= 1; off >>= 1) {
      float ov = __shfl_xor(v, off, 32);
      int oi = __shfl_xor(i, off, 32);
      if (ov > v || (ov == v && oi < i)) { v = ov; i = oi; }
    }
    if (n == 0) {
      int r = rl + j;
      pval[(size_t)r * NCHUNK + chunk] = v;
      pidx[(size_t)r * NCHUNK + chunk] = i;
    }
  }
}

// ---------------- final argmax combine + replacement gather ----------------
__global__ __launch_bounds__(128) void final_kernel(
    const float* __restrict__ pval, const int* __restrict__ pidx,
    const float* __restrict__ item_table,
    float* __restrict__ out_emb, float* __restrict__ out_idx)
{
  __shared__ float v[128];
  __shared__ int ix[128];
  const int row = blockIdx.x, t = threadIdx.x;
  float vv = -__builtin_inff(); int ii = 0x7fffffff;
  if (t < NCHUNK) { vv = pval[(size_t)row * NCHUNK + t]; ii = pidx[(size_t)row * NCHUNK + t]; }
  v[t] = vv; ix[t] = ii;
  __syncthreads();
  for (int s = 64; s > 0; s >>= 1) {
    if (t < s) {
      if (v[t + s] > v[t] || (v[t + s] == v[t] && ix[t + s] < ix[t])) {
        v[t] = v[t + s]; ix[t] = ix[t + s];
      }
    }
    __syncthreads();
  }
  const int bestIdx = ix[0];
  if (t < D_) out_emb[(size_t)row * D_ + t] = item_table[(size_t)bestIdx * D_ + t];
  if (t == 0) out_idx[row] = (float)bestIdx;
}

// ---------------- launcher ----------------
extern "C" void kernel_launch(void* const* d_in, const int* in_sizes, int n_in,
                              void* d_out, int out_size, void* d_ws, size_t ws_size,
                              hipStream_t stream) {
  const int* user_idx        = (const int*)d_in[0];
  const int* inter_idx       = (const int*)d_in[1];
  const float* priv          = (const float*)d_in[2];
  const unsigned char* mask  = (const unsigned char*)d_in[3];
  const float* user_table    = (const float*)d_in[4];
  const float* item_table    = (const float*)d_in[5];
  const float* w1            = (const float*)d_in[6];
  const float* b1            = (const float*)d_in[7];
  const float* w2            = (const float*)d_in[8];
  const float* b2            = (const float*)d_in[9];
  const float* wg            = (const float*)d_in[10];
  const float* bg            = (const float*)d_in[11];
  const float* gumbel        = (const float*)d_in[12];

  float* out = (float*)d_out;
  float* out_user = out;                    // [32,64]        = 2048
  float* out_item = out + 2048;             // [32,50,64]     = 102400
  float* out_dist = out + 104448;           // [32,50,50000]  = 80000000
  float* out_attp = out + 80104448;         // [32,50]        = 1600
  float* out_remb = out + 80106048;         // [32,50,64]     = 102400
  float* out_ridx = out + 80208448;         // [32,50]        = 1600

  char* ws = (char*)d_ws;
  float* ws_latent  = (float*)(ws);                 // 409600 B
  float* ws_att     = (float*)(ws + 409600);        // 6400 B
  unsigned* ws_bits = (unsigned*)(ws + 416000);     // 200704 B
  float* ws_pval    = (float*)(ws + 616704);        // 800000 B
  int* ws_pidx      = (int*)(ws + 1416704);         // 800000 B (total ~2.2 MB)

  bits_zero_kernel<<<dim3((B_ * BITS_W + 255) / 256), 256, 0, stream>>>(ws_bits);
  bits_scatter_kernel<<<dim3(7), 256, 0, stream>>>(inter_idx, mask, ws_bits);
  mlp_kernel<<<dim3(ROWS), 64, 0, stream>>>(user_idx, inter_idx, priv, mask,
                                            user_table, item_table, w1, b1, w2, b2,
                                            wg, bg, out_user, out_item,
                                            ws_latent, ws_att);
  softmax_kernel<<<dim3(B_), 64, 0, stream>>>(ws_att, out_attp);
  gemm_kernel<<<dim3(25, NCHUNK), 128, 0, stream>>>(item_table, inter_idx, ws_latent,
                                                    gumbel, ws_bits, out_dist,
                                                    ws_pval, ws_pidx);
  final_kernel<<<dim3(ROWS), 128, 0, stream>>>(ws_pval, ws_pidx, item_table,
                                               out_remb, out_ridx);
}